// a_mean_op_52793738003171
// MI455X (gfx1250) — compile-verified
//
#include <hip/hip_runtime.h>

typedef __attribute__((ext_vector_type(2))) float v2f;
typedef __attribute__((ext_vector_type(8))) float v8f;

#define FEAT 128

// ---------------------------------------------------------------------------
// hr = relu(h @ W.T + b) via V_WMMA_F32_16X16X4_F32.
// One wave32 owns a 16-row tile and computes all 8 column tiles (16x128 out),
// so h is streamed exactly once. K-loop: 128/4 = 32 WMMA steps per col tile.
//
// A (16x4 f32, ISA 7.12.2): lane l<16 -> M=l, K=k..k+1 ; lane l>=16 -> M=l-16,
// K=k+2..k+3  => per-lane float2 load h[row][k + 2*(l>>4)].
// B (4x16, B[k'][n] = W[n][k']): same striping => float2 load W[n][k + 2*(l>>4)].
// C/D (16x16 f32): VGPR j holds row = j + 8*(l>>4), col = l&15.
// ---------------------------------------------------------------------------
__global__ __launch_bounds__(128) void gnn_gemm_relu(
    const float* __restrict__ h, const float* __restrict__ W,
    const float* __restrict__ bias, float* __restrict__ hr, int n_nodes)
{
    const int lane     = threadIdx.x & 31;
    const int gwave    = blockIdx.x * (blockDim.x >> 5) + (threadIdx.x >> 5);
    const int n_tiles  = (n_nodes + 15) >> 4;
    if (gwave >= n_tiles) return;

    const int row0     = gwave << 4;
    const int lane_lo  = lane & 15;
    const int hi_half  = lane >> 4;      // 0 or 1
    const int k_off    = hi_half * 2;    // 0 or 2

    int arow = row0 + lane_lo;
    if (arow >= n_nodes) arow = n_nodes - 1;   // clamp: keeps EXEC all-ones for WMMA

    const float* __restrict__ hA = h + (size_t)arow * FEAT + k_off;

    v8f acc[8] = {};   // 8 col tiles of 16x16 f32 accumulators

#pragma unroll 2
    for (int k = 0; k < FEAT; k += 4) {
        const v2f a = *(const v2f*)(hA + k);
#pragma unroll
        for (int c = 0; c < 8; ++c) {
            const int n = (c << 4) + lane_lo;
            const v2f bb = *(const v2f*)(W + (size_t)n * FEAT + k + k_off);
            acc[c] = __builtin_amdgcn_wmma_f32_16x16x4_f32(
                false, a, false, bb, (short)0, acc[c], false, false);
        }
    }

    // Epilogue: bias + ReLU, scatter C layout back to row-major hr.
    // Bounds check hoisted to a *scalar* per-tile condition so the common
    // (full-tile) path is 64 unconditional, clause-able global_store_b32.
    float* __restrict__ hrRow = hr + (size_t)(row0 + hi_half * 8) * FEAT + lane_lo;
    if (row0 + 16 <= n_nodes) {
#pragma unroll
        for (int c = 0; c < 8; ++c) {
            const float bv = bias[(c << 4) + lane_lo];
#pragma unroll
            for (int j = 0; j < 8; ++j) {
                const float v = acc[c][j] + bv;
                hrRow[(size_t)j * FEAT + (c << 4)] = v > 0.0f ? v : 0.0f;
            }
        }
    } else {
        // Tail tile (only if n_nodes % 16 != 0): per-row guard.
#pragma unroll
        for (int c = 0; c < 8; ++c) {
            const float bv = bias[(c << 4) + lane_lo];
#pragma unroll
            for (int j = 0; j < 8; ++j) {
                const int row = row0 + j + hi_half * 8;
                if (row < n_nodes) {
                    const float v = acc[c][j] + bv;
                    hr[(size_t)row * FEAT + (c << 4) + lane_lo] = v > 0.0f ? v : 0.0f;
                }
            }
        }
    }
}

// Zero the output accumulator and the degree array.
__global__ void gnn_zero(float* __restrict__ out, float* __restrict__ deg, int n_nodes)
{
    const size_t i     = (size_t)blockIdx.x * blockDim.x + threadIdx.x;
    const size_t total = (size_t)n_nodes * FEAT;
    if (i < total)            out[i] = 0.0f;
    if (i < (size_t)n_nodes)  deg[i] = 0.0f;
}

// In-degree histogram: 1 atomic per edge (deg is 0.4 MB -> L2-resident).
__global__ void gnn_degree(const int* __restrict__ dst, float* __restrict__ deg, int n_edges)
{
    const int e = blockIdx.x * blockDim.x + threadIdx.x;
    if (e < n_edges) atomicAdd(&deg[dst[e]], 1.0f);
}

// One wave per edge: 16B vector gather of hr[src] + 4 f32 atomics into out[dst].
// src/dst are wave-uniform -> readfirstlane forces the scalar path.
__global__ __launch_bounds__(256) void gnn_scatter(
    const float* __restrict__ hr, const int* __restrict__ src,
    const int* __restrict__ dst, float* __restrict__ out, int n_edges)
{
    const int lane = threadIdx.x & 31;
    const int e    = (int)((blockIdx.x * blockDim.x + threadIdx.x) >> 5);
    if (e >= n_edges) return;

    const int s = __builtin_amdgcn_readfirstlane(src[e]);
    const int d = __builtin_amdgcn_readfirstlane(dst[e]);

    const float4 v = *((const float4*)(hr + (size_t)s * FEAT) + lane);
    float* pd = out + (size_t)d * FEAT + lane * 4;
    atomicAdd(pd + 0, v.x);
    atomicAdd(pd + 1, v.y);
    atomicAdd(pd + 2, v.z);
    atomicAdd(pd + 3, v.w);
}

// out = deg>0 ? msg_sum/deg : hr   (matches jnp.where(deg>0, sum/max(deg,1), hr))
__global__ void gnn_finalize(const float* __restrict__ hr, const float* __restrict__ deg,
                             float* __restrict__ out, int n_nodes)
{
    const size_t i     = (size_t)blockIdx.x * blockDim.x + threadIdx.x;
    const size_t total = (size_t)n_nodes * FEAT;
    if (i >= total) return;
    const int   node = (int)(i >> 7);           // FEAT == 128
    const float dg   = deg[node];
    out[i] = dg > 0.0f ? out[i] / dg : hr[i];
}

extern "C" void kernel_launch(void* const* d_in, const int* in_sizes, int n_in,
                              void* d_out, int out_size, void* d_ws, size_t ws_size,
                              hipStream_t stream)
{
    // setup_inputs order: h, h_in (unused), src, dst, W, b
    const float* h    = (const float*)d_in[0];
    const int*   src  = (const int*)  d_in[2];
    const int*   dst  = (const int*)  d_in[3];
    const float* W    = (const float*)d_in[4];
    const float* bias = (const float*)d_in[5];
    float*       out  = (float*)d_out;

    const int n_nodes = in_sizes[0] / FEAT;
    const int n_edges = in_sizes[2];

    // Workspace: hr [n_nodes*FEAT floats] then deg [n_nodes floats] (~51.6 MB).
    float* hr  = (float*)d_ws;
    float* deg = hr + (size_t)n_nodes * FEAT;

    const size_t total = (size_t)n_nodes * FEAT;

    {   // zero out + deg
        const int threads = 256;
        const int blocks  = (int)((total + threads - 1) / threads);
        gnn_zero<<<blocks, threads, 0, stream>>>(out, deg, n_nodes);
    }
    {   // WMMA GEMM + bias + ReLU -> hr
        const int n_tiles = (n_nodes + 15) / 16;
        const int wpb     = 4;                     // 128 threads = 4 waves
        const int blocks  = (n_tiles + wpb - 1) / wpb;
        gnn_gemm_relu<<<blocks, 128, 0, stream>>>(h, W, bias, hr, n_nodes);
    }
    {   // degree histogram
        const int threads = 256;
        const int blocks  = (n_edges + threads - 1) / threads;
        gnn_degree<<<blocks, threads, 0, stream>>>(dst, deg, n_edges);
    }
    {   // edge gather + scatter-add (one wave per edge)
        const int wpb    = 8;                      // 256 threads = 8 waves
        const int blocks = (n_edges + wpb - 1) / wpb;
        gnn_scatter<<<blocks, 256, 0, stream>>>(hr, src, dst, out, n_edges);
    }
    {   // mean + zero-in-degree fallback
        const int threads = 256;
        const int blocks  = (int)((total + threads - 1) / threads);
        gnn_finalize<<<blocks, threads, 0, stream>>>(hr, deg, out, n_nodes);
    }
}